// Language_16329465660246
// MI455X (gfx1250) — compile-verified
//
#include <hip/hip_runtime.h>

// CDNA5 / gfx1250 implementation of Language.compute_probs fixed-point:
//   probs <- nullary + w0*SP_app(probs) + w1*SP_comp(probs) + w2*SP_join(probs)
// Edge passes are streaming-bandwidth bound (288MB indices/step); gathers and
// scatter-atomics hit the 192MB L2. The weighted combine is done with
// V_WMMA_F32_16X16X4_F32 (A = [S0|S1|S2|nullary], B = [w0;w1;w2;1]).

typedef __attribute__((ext_vector_type(4))) int   i4;
typedef __attribute__((ext_vector_type(2))) float v2f;
typedef __attribute__((ext_vector_type(8))) float v8f;

// Guaranteed-native no-return fp32 atomic add (L2 atomic units, device scope).
// Avoids any risk of a CAS-loop expansion of the HIP atomic builtin.
__device__ __forceinline__ void atomic_add_f32_noret(float* p, float val) {
#if defined(__gfx1250__)
    asm volatile("global_atomic_add_f32 %0, %1, off scope:SCOPE_DEV"
                 :
                 : "v"((unsigned long long)(uintptr_t)p), "v"(val)
                 : "memory");
#else
    __hip_atomic_fetch_add(p, val, __ATOMIC_RELAXED, __HIP_MEMORY_SCOPE_AGENT);
#endif
}

// ---------------------------------------------------------------------------
// Zero the three per-table accumulators.
// ---------------------------------------------------------------------------
__global__ void zero3_kernel(float* __restrict__ a, float* __restrict__ b,
                             float* __restrict__ c, int n) {
    int i  = blockIdx.x * blockDim.x + threadIdx.x;
    int st = gridDim.x * blockDim.x;
    for (; i < n; i += st) { a[i] = 0.0f; b[i] = 0.0f; c[i] = 0.0f; }
}

// ---------------------------------------------------------------------------
// Edge pass: acc[v] += x[l]*x[r] over 8M triples (4 edges/thread, b128 loads).
// NT=true marks the index streams non-temporal so they do not thrash L2
// (app+comp tables = 192MB are left cacheable; join streams past).
// ---------------------------------------------------------------------------
template<bool NT>
__global__ __launch_bounds__(256) void edge_accum_kernel(
    const float* __restrict__ x,
    const i4* __restrict__ l4, const i4* __restrict__ r4,
    const i4* __restrict__ v4,
    float* __restrict__ acc, int nquads) {
    int i  = blockIdx.x * blockDim.x + threadIdx.x;
    int st = gridDim.x * blockDim.x;
    for (; i < nquads; i += st) {
        i4 l, r, v;
        if (NT) {
            l = __builtin_nontemporal_load(l4 + i);
            r = __builtin_nontemporal_load(r4 + i);
            v = __builtin_nontemporal_load(v4 + i);
        } else {
            l = l4[i]; r = r4[i]; v = v4[i];
        }
        float p0 = x[l.x] * x[r.x];
        float p1 = x[l.y] * x[r.y];
        float p2 = x[l.z] * x[r.z];
        float p3 = x[l.w] * x[r.w];
        atomic_add_f32_noret(acc + v.x, p0);
        atomic_add_f32_noret(acc + v.y, p1);
        atomic_add_f32_noret(acc + v.z, p2);
        atomic_add_f32_noret(acc + v.w, p3);
    }
}

// ---------------------------------------------------------------------------
// Weighted combine via WMMA:  dst = nullary + w0*s0 + w1*s1 + w2*s2.
// A (16x4 f32): lanes 0-15 hold {K0=s0, K1=s1} for M=lane; lanes 16-31 hold
// {K2=s2, K3=nullary} for M=lane-16.  B (4x16): column n = [w0,w1,w2,1].
// D: VGPR j, lane<16 -> M=j, N=lane; lane>=16 -> M=j+8. All N columns equal;
// lanes with N==0 write the 16 results.
// ---------------------------------------------------------------------------
__global__ __launch_bounds__(256) void combine_wmma_kernel(
    const float* __restrict__ s0, const float* __restrict__ s1,
    const float* __restrict__ s2, const float* __restrict__ nul,
    const float* __restrict__ w,  float* __restrict__ dst,
    int n_tiles, int n_total) {
    int tid   = blockIdx.x * blockDim.x + threadIdx.x;
    int lane  = threadIdx.x & 31;
    int wave  = tid >> 5;
    int nwave = (gridDim.x * blockDim.x) >> 5;

    bool lo = (lane < 16);
    int  m  = lane & 15;
    // cndmask-style selects: no EXEC divergence before the WMMA.
    const float* pax = lo ? s0 : s2;
    const float* pay = lo ? s1 : nul;
    v2f b;
    b.x = lo ? w[0] : w[2];
    b.y = lo ? w[1] : 1.0f;

    for (int t = wave; t < n_tiles; t += nwave) {   // wave-uniform bound
        int base = t * 16;
        v2f a;
        a.x = pax[base + m];
        a.y = pay[base + m];
        v8f c = {};
        c = __builtin_amdgcn_wmma_f32_16x16x4_f32(
                /*neg_a=*/false, a, /*neg_b=*/false, b,
                /*c_mod=*/(short)0, c, /*reuse_a=*/false, /*reuse_b=*/false);
        if ((lane & 15) == 0) {
            int mbase = base + ((lane >> 4) << 3);
#pragma unroll
            for (int j = 0; j < 8; ++j) dst[mbase + j] = c[j];
        }
    }

    // Scalar tail for n_total not divisible by 16 (none for N_OBS=500000).
    int st = gridDim.x * blockDim.x;
    for (int idx = n_tiles * 16 + tid; idx < n_total; idx += st)
        dst[idx] = nul[idx] + w[0] * s0[idx] + w[1] * s1[idx] + w[2] * s2[idx];
}

// ---------------------------------------------------------------------------
extern "C" void kernel_launch(void* const* d_in, const int* in_sizes, int n_in,
                              void* d_out, int out_size, void* d_ws, size_t ws_size,
                              hipStream_t stream) {
    const float* nullary = (const float*)d_in[0];
    const float* fnw     = (const float*)d_in[1];
    const int* app_l  = (const int*)d_in[2];
    const int* app_r  = (const int*)d_in[3];
    const int* app_v  = (const int*)d_in[4];
    const int* comp_l = (const int*)d_in[5];
    const int* comp_r = (const int*)d_in[6];
    const int* comp_v = (const int*)d_in[7];
    const int* join_l = (const int*)d_in[8];
    const int* join_r = (const int*)d_in[9];
    const int* join_v = (const int*)d_in[10];

    const int n_obs   = in_sizes[0];     // 500000
    const int n_edges = in_sizes[2];     // 8000000
    const int nquads  = n_edges / 4;
    const int n_tiles = n_obs / 16;

    float* probs = (float*)d_ws;         // 2MB
    float* acc0  = probs + n_obs;        // 2MB each
    float* acc1  = acc0 + n_obs;
    float* acc2  = acc1 + n_obs;

    // probs <- nullary (reference init). D2D async copy is capture-safe.
    hipMemcpyAsync(probs, nullary, (size_t)n_obs * sizeof(float),
                   hipMemcpyDeviceToDevice, stream);

    const int eblocks = (nquads + 255) / 256;
    const int zblocks = (n_obs + 255) / 256;
    const int cblocks = (n_tiles + 7) / 8;     // 8 waves/block, 1 tile/wave min
    const int NSTEPS  = 10;

    for (int s = 0; s < NSTEPS; ++s) {
        zero3_kernel<<<zblocks, 256, 0, stream>>>(acc0, acc1, acc2, n_obs);
        edge_accum_kernel<false><<<eblocks, 256, 0, stream>>>(
            probs, (const i4*)app_l, (const i4*)app_r, (const i4*)app_v, acc0, nquads);
        edge_accum_kernel<false><<<eblocks, 256, 0, stream>>>(
            probs, (const i4*)comp_l, (const i4*)comp_r, (const i4*)comp_v, acc1, nquads);
        edge_accum_kernel<true><<<eblocks, 256, 0, stream>>>(
            probs, (const i4*)join_l, (const i4*)join_r, (const i4*)join_v, acc2, nquads);
        float* dst = (s == NSTEPS - 1) ? (float*)d_out : probs;
        combine_wmma_kernel<<<cblocks, 256, 0, stream>>>(
            acc0, acc1, acc2, nullary, fnw, dst, n_tiles, n_obs);
    }
}